// GMP_27075473834662
// MI455X (gfx1250) — compile-verified
//
#include <hip/hip_runtime.h>

typedef __attribute__((ext_vector_type(2))) float v2f;
typedef __attribute__((ext_vector_type(4))) float v4f;
typedef __attribute__((ext_vector_type(8))) float v8f;

namespace {
constexpr int Tlen  = 8192;
constexpr int Bsz   = 16;
constexpr int Lc    = 13;
constexpr int PPAD  = Tlen + 64;   // halo-padded power plane length, offset +24
constexpr int POFF  = 24;
constexpr int NTILE = 513;         // 16-row v-tiles covering v in [0, T+11]
constexpr int CROWS = NTILE * 16;  // 8208

// contraction index j = 4*m + k  (k=3 is a zero pad slot -> offset == index)
constexpr int KJ12 = 52;           // m = 0..12 (m==12 row holds A_kl)
constexpr int KJ3  = 48;           // m = 0..11

// workspace layout (float offsets)
constexpr size_t P_OFF   = 0;                                   // Bsz*PPAD*4 interleaved {a,a3,a5,0}
constexpr size_t BP_OFF  = P_OFF + (size_t)Bsz * PPAD * 4;      // 16*KJ12  B~ transposed [col][j]
constexpr size_t CP_OFF  = BP_OFF + 16 * KJ12;                  // 16*KJ3   C  transposed [col][j]
constexpr size_t C12_OFF = CP_OFF + 16 * KJ3;                   // Bsz*CROWS*16
constexpr size_t C3_OFF  = C12_OFF + (size_t)Bsz * CROWS * 16;  // Bsz*CROWS*16
} // namespace

// Stage 0: pack coefficients as B-operand tiles [col=l][j=4m+k]; zero pads at k==3, col>=13.
__global__ void gmp_pack(const float* __restrict__ Akl,
                         const float* __restrict__ Bklm,
                         const float* __restrict__ Cklm,
                         float* __restrict__ ws) {
  int i = blockIdx.x * blockDim.x + threadIdx.x;
  float* Bp = ws + BP_OFF;
  float* Cp = ws + CP_OFF;
  if (i < 16 * KJ12) {
    int col = i / KJ12, j = i - col * KJ12;
    int m = j >> 2, k = j & 3;
    float v = 0.0f;
    if (col < Lc && k < 3)
      v = (m < 12) ? Bklm[(col * 12 + m) * 3 + k]   // B_klm[l][m][k]
                   : Akl[col * 3 + k];              // A_kl folded in at m==12
    Bp[i] = v;
  } else if (i < 16 * KJ12 + 16 * KJ3) {
    int i2 = i - 16 * KJ12;
    int col = i2 / KJ3, j = i2 - col * KJ3;
    int m = j >> 2, k = j & 3;
    Cp[i2] = (col < Lc && k < 3) ? Cklm[(col * 12 + m) * 3 + k] : 0.0f;
  }
}

// Stage 1: |x| powers, interleaved stride-4 with halo (mod-T wrap resolved here).
__global__ void gmp_powers(const float* __restrict__ x, float* __restrict__ ws) {
  int i = blockIdx.x * blockDim.x + threadIdx.x;
  if (i >= Bsz * PPAD) return;
  int b = i / PPAD, up = i - b * PPAD;
  int u = up - POFF;
  int idx = (u < 0) ? u + Tlen : ((u >= Tlen) ? u - Tlen : u);
  v2f xc = ((const v2f*)x)[(size_t)b * Tlen + idx];
  float a  = sqrtf(xc.x * xc.x + xc.y * xc.y);
  float a2 = a * a;
  v4f p = {a, a * a2, a * a2 * a2, 0.0f};
  ((v4f*)(ws + P_OFF))[(size_t)b * PPAD + up] = p;
}

// Stage 2: WMMA GEMMs, one wave per 16-row v-tile.
//   c12'[v,l]: K=52 -> 13x v_wmma_f32_16x16x4_f32 ; c3'[v,l]: K=48 -> 12x.
// Contraction index == float offset: all operand loads are one base + immediate-offset v2f.
__global__ void __launch_bounds__(128) gmp_wmma(float* __restrict__ ws) {
  const int wave = blockIdx.x * 4 + (threadIdx.x >> 5);
  if (wave >= Bsz * NTILE) return;           // uniform per wave: EXEC stays all-1s
  const int lane = threadIdx.x & 31;
  const int row  = lane & 15;                // A row / B,D column
  const int kh   = lane >> 4;                // lane half: K pair select / D row half
  const int b    = wave / NTILE;
  const int v0   = (wave - b * NTILE) * 16;

  // A element (row, j) lives at Abase + j floats (j = 4m+k, u-stride = 4 floats).
  const float* Abase = ws + P_OFF + ((size_t)b * PPAD + v0 + row) * 4 + 2 * kh;
  const float* Bbase = ws + BP_OFF + row * KJ12 + 2 * kh;
  const float* Cbase = ws + CP_OFF + row * KJ3 + 2 * kh;

  v8f acc12 = {};
  v8f acc3  = {};

#pragma unroll
  for (int s = 0; s < KJ12 / 4; ++s) {       // c12': u index = v0+row+m-24, halo offset cancels
    v2f a  = *(const v2f*)(Abase + 4 * s);
    v2f bm = *(const v2f*)(Bbase + 4 * s);
    acc12 = __builtin_amdgcn_wmma_f32_16x16x4_f32(false, a, false, bm,
                                                  (short)0, acc12, false, false);
  }
#pragma unroll
  for (int s = 0; s < KJ3 / 4; ++s) {        // c3': u index = v0+row+m  (+POFF*4 floats)
    v2f a  = *(const v2f*)(Abase + POFF * 4 + 4 * s);
    v2f bm = *(const v2f*)(Cbase + 4 * s);
    acc3 = __builtin_amdgcn_wmma_f32_16x16x4_f32(false, a, false, bm,
                                                 (short)0, acc3, false, false);
  }

  float* C12 = ws + C12_OFF;
  float* C3  = ws + C3_OFF;
#pragma unroll
  for (int r = 0; r < 8; ++r) {              // D: VGPR r -> rows r (lanes 0-15), r+8 (16-31)
    const int u = v0 + r + 8 * kh;
    C12[((size_t)b * CROWS + u) * 16 + row] = acc12[r];
    C3 [((size_t)b * CROWS + u) * 16 + row] = acc3[r];
  }
}

// Stage 3: diagonal gather + complex taps:
//   s[t] = sum_l x[(t+l-12)%T]*c12'[t+l,l] + x[(t+l+12)%T]*c3'[t+l,l]
__global__ void gmp_combine(const float* __restrict__ x, const float* __restrict__ ws,
                            float* __restrict__ out) {
  int i = blockIdx.x * blockDim.x + threadIdx.x;
  if (i >= Bsz * Tlen) return;
  int b = i / Tlen, t = i - b * Tlen;
  const float* C12 = ws + C12_OFF;
  const float* C3  = ws + C3_OFF;
  const v2f*   xv  = (const v2f*)x;
  float sr = 0.0f, si = 0.0f;
#pragma unroll
  for (int l = 0; l < Lc; ++l) {
    int v  = t + l;
    int i1 = t + l - 12; i1 = (i1 < 0) ? i1 + Tlen : ((i1 >= Tlen) ? i1 - Tlen : i1);
    int i3 = t + l + 12; if (i3 >= Tlen) i3 -= Tlen;
    float c12v = C12[((size_t)b * CROWS + v) * 16 + l];
    float c3v  = C3 [((size_t)b * CROWS + v) * 16 + l];
    v2f x1 = xv[(size_t)b * Tlen + i1];
    v2f x3 = xv[(size_t)b * Tlen + i3];
    sr = fmaf(x1.x, c12v, sr); si = fmaf(x1.y, c12v, si);
    sr = fmaf(x3.x, c3v, sr);  si = fmaf(x3.y, c3v, si);
  }
  out[(size_t)i * 2 + 0] = sr;
  out[(size_t)i * 2 + 1] = si;
}

extern "C" void kernel_launch(void* const* d_in, const int* in_sizes, int n_in,
                              void* d_out, int out_size, void* d_ws, size_t ws_size,
                              hipStream_t stream) {
  const float* x    = (const float*)d_in[0];
  // d_in[1] = h_0 (unused by reference output)
  const float* Akl  = (const float*)d_in[2];
  const float* Bklm = (const float*)d_in[3];
  const float* Cklm = (const float*)d_in[4];
  float* ws  = (float*)d_ws;
  float* out = (float*)d_out;

  gmp_pack   <<<(16 * (KJ12 + KJ3) + 255) / 256, 256, 0, stream>>>(Akl, Bklm, Cklm, ws);
  gmp_powers <<<(Bsz * PPAD + 255) / 256, 256, 0, stream>>>(x, ws);
  gmp_wmma   <<<(Bsz * NTILE + 3) / 4, 128, 0, stream>>>(ws);
  gmp_combine<<<(Bsz * Tlen + 255) / 256, 256, 0, stream>>>(x, ws, out);
}